// GNNLSTM_20186346291942
// MI455X (gfx1250) — compile-verified
//
#include <hip/hip_runtime.h>
#include <hip/hip_bf16.h>

typedef __attribute__((ext_vector_type(2))) float v2f;
typedef __attribute__((ext_vector_type(8))) float v8f;

#define NN_NODES   100000
#define NN_EDGES   1600000
#define IN_DIM     128
#define HID        128
#define NCLS       10
#define NGRAPH     64

// ---------------------------------------------------------------------------
// WMMA f32 GEMM: C[M,N] = A[M,K] * B  (K must be 128 here, M % 32 == 0).
// TRANSB=0: B is [K,N] row-major.  TRANSB=1: B is [N,K] row-major (i.e. B^T).
// Block = 256 threads = 8 waves; block computes a 32 x 128 strip of C.
// Each wave keeps TWO 16x16 f32 accumulators (rows m0..m0+15 / m0+16..m0+31)
// so every B fragment is reused by two v_wmma_f32_16x16x4_f32 ops.
// A-tile (32 x 128 = 16 KB, contiguous in memory) is staged into LDS with
// CDNA5 async DMA (global_load_async_to_lds_b128, ASYNCcnt-tracked).
// grid.x = M/32, grid.y = N/128.
// Fragment layout (per CDNA5 ISA 7.12.2):
//   A 16x4 f32 : lanes 0-15 hold (K=k,k+1), lanes 16-31 hold (K=k+2,k+3)
//   B 4x16 f32 : same half-wave K split, row striped across lanes (N=lane%16)
//   C 16x16    : VGPR r -> row r (lanes 0-15) / row r+8 (lanes 16-31)
// ---------------------------------------------------------------------------
template <bool TRANSB>
__global__ __launch_bounds__(256)
void gemm_wmma_f32(const float* __restrict__ A, const float* __restrict__ B,
                   float* __restrict__ C, int M, int N, int K) {
    __shared__ float Atile[32 * 128];

    const int tid  = threadIdx.x;
    const int m0   = blockIdx.x * 32;
    const int wave = tid >> 5;
    const int lane = tid & 31;          // wave32
    const int half = lane >> 4;
    const int lr   = lane & 15;
    const int n0   = blockIdx.y * 128 + wave * 16;
    const int n    = n0 + lr;

    // --- async-DMA stage of the contiguous 32x128 A tile into LDS ----------
    // Generic LDS addr[31:0] == LDS byte offset (ISA 10.2 aperture decode).
    const unsigned abase = (unsigned)(uintptr_t)&Atile[0];
    const float*   Ablk  = A + (size_t)m0 * K;      // tile is contiguous (full rows)
    for (int i = tid * 4; i < 32 * 128; i += 256 * 4) {
        unsigned     ldsoff = abase + (unsigned)(i * 4);
        const float* gsrc   = Ablk + i;
        asm volatile("global_load_async_to_lds_b128 %0, %1, off"
                     :: "v"(ldsoff), "v"(gsrc) : "memory");
    }
    asm volatile("s_wait_asynccnt 0x0" ::: "memory");
    __syncthreads();

    v8f acc0 = {}, acc1 = {};
    for (int k = 0; k < K; k += 4) {
        const int ka = k + half * 2;
        v2f a0, a1, b;
        a0.x = Atile[lr * 128 + ka];
        a0.y = Atile[lr * 128 + ka + 1];
        a1.x = Atile[(16 + lr) * 128 + ka];
        a1.y = Atile[(16 + lr) * 128 + ka + 1];
        if (TRANSB) {
            b.x = B[(size_t)n * K + ka];
            b.y = B[(size_t)n * K + ka + 1];
        } else {
            b.x = B[(size_t)ka * N + n];
            b.y = B[(size_t)(ka + 1) * N + n];
        }
        acc0 = __builtin_amdgcn_wmma_f32_16x16x4_f32(
            false, a0, false, b, (short)0, acc0, false, false);
        acc1 = __builtin_amdgcn_wmma_f32_16x16x4_f32(
            false, a1, false, b, (short)0, acc1, false, false);
    }

    const int row0 = m0 + half * 8;
#pragma unroll
    for (int r = 0; r < 8; ++r) {
        C[(size_t)(row0 + r) * N + n]      = acc0[r];
        C[(size_t)(row0 + 16 + r) * N + n] = acc1[r];
    }
}

// ---------------------------------------------------------------------------
// deg[c] += 1 for each edge target c  (deg buffer pre-zeroed)
// ---------------------------------------------------------------------------
__global__ void deg_count(const int* __restrict__ ei, float* __restrict__ deg, int E) {
    int e = blockIdx.x * blockDim.x + threadIdx.x;
    if (e < E) atomicAdd(&deg[ei[E + e]], 1.0f);
}

// dinv = rsqrt(deg + 1)  (in place)
__global__ void dinv_k(float* __restrict__ deg, int n) {
    int i = blockIdx.x * blockDim.x + threadIdx.x;
    if (i < n) deg[i] = rsqrtf(deg[i] + 1.0f);
}

// agg = xt * dinv^2   (self-loop term; also initializes agg)
__global__ void selfloop_init(const float* __restrict__ xt,
                              const float* __restrict__ dinv,
                              float* __restrict__ agg) {
    int i = blockIdx.x * blockDim.x + threadIdx.x;   // over N_NODES*HID
    float d = dinv[i >> 7];
    agg[i] = xt[i] * d * d;
}

// One wave32 per edge; each lane moves a float4 slice of the 128-wide row.
__global__ __launch_bounds__(256)
void edge_scatter(const int* __restrict__ ei, const float* __restrict__ xt,
                  const float* __restrict__ dinv, float* __restrict__ agg, int E) {
    int gtid = blockIdx.x * 256 + threadIdx.x;
    int e    = gtid >> 5;
    int lane = gtid & 31;
    if (e >= E) return;
    int r = ei[e];
    int c = ei[E + e];
    float norm = dinv[r] * dinv[c];
    float4 v = *((const float4*)(xt + (size_t)r * HID) + lane);
    float* dst = agg + (size_t)c * HID + lane * 4;
    atomicAdd(dst + 0, v.x * norm);
    atomicAdd(dst + 1, v.y * norm);
    atomicAdd(dst + 2, v.z * norm);
    atomicAdd(dst + 3, v.w * norm);
}

// relu(agg + b) then atomic mean-pool accumulation per graph
__global__ __launch_bounds__(128)
void relu_pool(const float* __restrict__ agg, const float* __restrict__ b_gcn,
               const int* __restrict__ batch, float* __restrict__ psum,
               float* __restrict__ pcnt) {
    int node = blockIdx.x;
    int h    = threadIdx.x;
    float v  = fmaxf(agg[(size_t)node * HID + h] + b_gcn[h], 0.0f);
    int g    = batch[node];
    atomicAdd(&psum[g * HID + h], v);
    if (h == 0) atomicAdd(&pcnt[g], 1.0f);
}

__global__ void pool_div(const float* __restrict__ psum, const float* __restrict__ pcnt,
                         float* __restrict__ pooled) {
    int i = blockIdx.x * blockDim.x + threadIdx.x;   // NGRAPH*HID = 8192
    pooled[i] = psum[i] / fmaxf(pcnt[i >> 7], 1.0f);
}

// ---------------------------------------------------------------------------
// Sequential LSTM, single block of 512 threads (one thread per gate unit).
// xw[t][j] = (pooled @ w_ih^T)[t][j] precomputed by WMMA GEMM.
// Gate order (torch): i, f, g, o.
// ---------------------------------------------------------------------------
__global__ __launch_bounds__(512)
void lstm_k(const float* __restrict__ xw, const float* __restrict__ w_hh,
            const float* __restrict__ b_ih, const float* __restrict__ b_hh,
            float* __restrict__ hs) {
    __shared__ float h[HID], c[HID], gates[4 * HID];
    int tid = threadIdx.x;
    if (tid < HID) { h[tid] = 0.0f; c[tid] = 0.0f; }
    __syncthreads();

    for (int t = 0; t < NGRAPH; ++t) {
        float acc = xw[t * 4 * HID + tid] + b_ih[tid] + b_hh[tid];
        const float* wr = w_hh + (size_t)tid * HID;
#pragma unroll 8
        for (int k = 0; k < HID; ++k) acc += h[k] * wr[k];
        gates[tid] = acc;
        __syncthreads();
        if (tid < HID) {
            float ig = 1.0f / (1.0f + __expf(-gates[tid]));
            float fg = 1.0f / (1.0f + __expf(-gates[HID + tid]));
            float gg = tanhf(gates[2 * HID + tid]);
            float og = 1.0f / (1.0f + __expf(-gates[3 * HID + tid]));
            float cn = fg * c[tid] + ig * gg;
            float hn = og * tanhf(cn);
            c[tid] = cn;
            h[tid] = hn;
            hs[t * HID + tid] = hn;
        }
        __syncthreads();
    }
}

// out[t][cls] = hs[t] . W_fc[cls] + b_fc[cls]
__global__ void fc_k(const float* __restrict__ hs, const float* __restrict__ W_fc,
                     const float* __restrict__ b_fc, float* __restrict__ out) {
    int i = blockIdx.x * blockDim.x + threadIdx.x;
    if (i >= NGRAPH * NCLS) return;
    int t = i / NCLS, cls = i % NCLS;
    const float* hr = hs + t * HID;
    const float* wr = W_fc + cls * HID;
    float acc = b_fc[cls];
    for (int k = 0; k < HID; ++k) acc += hr[k] * wr[k];
    out[i] = acc;
}

// ---------------------------------------------------------------------------
extern "C" void kernel_launch(void* const* d_in, const int* in_sizes, int n_in,
                              void* d_out, int out_size, void* d_ws, size_t ws_size,
                              hipStream_t stream) {
    const float* x     = (const float*)d_in[0];
    const int*   ei    = (const int*)d_in[1];   // [2, E]
    const int*   batch = (const int*)d_in[2];
    const float* W_gcn = (const float*)d_in[3];
    const float* b_gcn = (const float*)d_in[4];
    const float* w_ih  = (const float*)d_in[5]; // [512,128]
    const float* w_hh  = (const float*)d_in[6]; // [512,128]
    const float* b_ih  = (const float*)d_in[7];
    const float* b_hh  = (const float*)d_in[8];
    const float* W_fc  = (const float*)d_in[9]; // [10,128]
    const float* b_fc  = (const float*)d_in[10];
    float* out = (float*)d_out;

    // workspace layout (floats)
    float* ws     = (float*)d_ws;
    float* deg    = ws;                   // 100000 (becomes dinv in-place)
    float* xt     = ws + 102400;          // 12.8M : x @ W_gcn
    float* agg    = xt + 12800000;        // 12.8M : normalized aggregation
    float* psum   = agg + 12800000;       // 8192
    float* pcnt   = psum + 8192;          // 64 (padded to 128)
    float* pooled = pcnt + 128;           // 8192
    float* xw     = pooled + 8192;        // 32768 : pooled @ w_ih^T
    float* hs     = xw + 32768;           // 8192

    hipMemsetAsync(deg, 0, 100000 * sizeof(float), stream);
    hipMemsetAsync(psum, 0, (8192 + 128) * sizeof(float), stream);

    deg_count<<<(NN_EDGES + 255) / 256, 256, 0, stream>>>(ei, deg, NN_EDGES);
    dinv_k<<<(NN_NODES + 255) / 256, 256, 0, stream>>>(deg, NN_NODES);

    // xt = x @ W_gcn   (M=100000, N=128, K=128)
    dim3 g1(NN_NODES / 32, 1);
    gemm_wmma_f32<false><<<g1, 256, 0, stream>>>(x, W_gcn, xt, NN_NODES, HID, IN_DIM);

    selfloop_init<<<(NN_NODES * HID) / 256, 256, 0, stream>>>(xt, deg, agg);
    edge_scatter<<<(NN_EDGES * 32) / 256, 256, 0, stream>>>(ei, xt, deg, agg, NN_EDGES);
    relu_pool<<<NN_NODES, 128, 0, stream>>>(agg, b_gcn, batch, psum, pcnt);
    pool_div<<<(NGRAPH * HID) / 256, 256, 0, stream>>>(psum, pcnt, pooled);

    // xw = pooled @ w_ih^T   (M=64, N=512, K=128)
    dim3 g2(NGRAPH / 32, (4 * HID) / 128);
    gemm_wmma_f32<true><<<g2, 256, 0, stream>>>(pooled, w_ih, xw, NGRAPH, 4 * HID, HID);

    lstm_k<<<1, 512, 0, stream>>>(xw, w_hh, b_ih, b_hh, hs);
    fc_k<<<(NGRAPH * NCLS + 255) / 256, 256, 0, stream>>>(hs, W_fc, b_fc, out);
}